// PerPixelChannelPermutation_26792005992692
// MI455X (gfx1250) — compile-verified
//
#include <hip/hip_runtime.h>
#include <hip/hip_bf16.h>
#include <stdint.h>

// ---------------------------------------------------------------------------
// Per-pixel random channel permutation, 16 x 2048 x 2048 f32.
//   out[k, i, j] = image[perm_{ij}[k], i, j],  perm = argsort(threefry uniforms)
// Memory roofline: 512 MB @ 23.3 TB/s ~ 22 us; actual bound is the bit-exact
// JAX threefry RNG (~70 VALU per 64-bit block, 32M blocks). One thread =
// pixel pair (p, p + 2M) sharing threefry blocks (JAX counter split), async
// global->LDS staging overlapped with RNG + Batcher sorting network.
// ---------------------------------------------------------------------------

namespace {
constexpr uint32_t kC        = 16;
constexpr uint32_t kW        = 2048;
constexpr uint32_t kH        = 2048;
constexpr uint32_t kPlane    = kW * kH;          // 4,194,304 pixels / channel
constexpr uint32_t kHalfPix  = kPlane / 2;       // 2,097,152
constexpr uint32_t kHalfElem = kHalfPix * kC;    // 33,554,432 = N/2 counter split
constexpr uint32_t kThreads  = 256;
constexpr uint32_t kLdsStride = 17;              // odd stride -> bank-conflict-free
}

// ---- CDNA5 async global->LDS copy (probe-confirmed builtin) ---------------
#define ASYNC_COPY_B32(gptr, lptr)                                            \
  __builtin_amdgcn_global_load_async_to_lds_b32(                              \
      (__attribute__((address_space(1))) int*)(uintptr_t)(gptr),              \
      (__attribute__((address_space(3))) int*)(uint32_t)(uintptr_t)(lptr),    \
      0, 0)

// ---- JAX threefry2x32, key = PRNGKey(1) = (k0=0, k1=1) --------------------
__device__ __forceinline__ void threefry2x32_key01(uint32_t x0, uint32_t x1,
                                                   uint32_t& o0, uint32_t& o1) {
  constexpr uint32_t ks1 = 1u;
  constexpr uint32_t ks2 = 0x1BD11BDBu;  // 0 ^ 1 ^ 0x1BD11BDA
  x0 += 0u;  x1 += ks1;
#define TF_ROUND(r)                                     \
  { x0 += x1; x1 = __builtin_rotateleft32(x1, (r)); x1 ^= x0; }
  TF_ROUND(13) TF_ROUND(15) TF_ROUND(26) TF_ROUND(6)
  x0 += ks1; x1 += ks2 + 1u;
  TF_ROUND(17) TF_ROUND(29) TF_ROUND(16) TF_ROUND(24)
  x0 += ks2; x1 += 0u + 2u;
  TF_ROUND(13) TF_ROUND(15) TF_ROUND(26) TF_ROUND(6)
  x0 += 0u;  x1 += ks1 + 3u;
  TF_ROUND(17) TF_ROUND(29) TF_ROUND(16) TF_ROUND(24)
  x0 += ks1; x1 += ks2 + 4u;
  TF_ROUND(13) TF_ROUND(15) TF_ROUND(26) TF_ROUND(6)
  x0 += ks2; x1 += 0u + 5u;
#undef TF_ROUND
  o0 = x0; o1 = x1;
}

// ---- Batcher odd-even mergesort, n=16, 63 compare-exchanges (ascending) ---
// Hardcoded pair list -> every index is a literal, arrays stay in VGPRs,
// each CE lowers to v_min_u32 + v_max_u32.
__device__ __forceinline__ void batcher16(uint32_t k[16]) {
#define CE(i, j)                                         \
  { uint32_t a = k[i], b = k[j];                         \
    k[i] = a < b ? a : b;  /* v_min_u32 */               \
    k[j] = a < b ? b : a;  /* v_max_u32 */ }
  CE(0,1)  CE(2,3)  CE(4,5)  CE(6,7)  CE(8,9)  CE(10,11) CE(12,13) CE(14,15)
  CE(0,2)  CE(1,3)  CE(4,6)  CE(5,7)  CE(8,10) CE(9,11)  CE(12,14) CE(13,15)
  CE(1,2)  CE(5,6)  CE(9,10) CE(13,14)
  CE(0,4)  CE(1,5)  CE(2,6)  CE(3,7)  CE(8,12) CE(9,13)  CE(10,14) CE(11,15)
  CE(2,4)  CE(3,5)  CE(10,12) CE(11,13)
  CE(1,2)  CE(3,4)  CE(5,6)  CE(9,10) CE(11,12) CE(13,14)
  CE(0,8)  CE(1,9)  CE(2,10) CE(3,11) CE(4,12) CE(5,13)  CE(6,14)  CE(7,15)
  CE(4,8)  CE(5,9)  CE(6,10) CE(7,11)
  CE(2,4)  CE(3,5)  CE(6,8)  CE(7,9)  CE(10,12) CE(11,13)
  CE(1,2)  CE(3,4)  CE(5,6)  CE(7,8)  CE(9,10) CE(11,12) CE(13,14)
#undef CE
}

__global__ __launch_bounds__(kThreads)
void PerPixelChannelPermutation_kernel(const float* __restrict__ img,
                                       float* __restrict__ out) {
  const uint32_t tid = threadIdx.x;
  const uint32_t p   = blockIdx.x * kThreads + tid;  // pixel in [0, kHalfPix)
  const uint32_t q   = p + kHalfPix;                 // paired pixel

  // 512 pixel slots * 17 floats (16 channels + 1 pad) = 34,816 B LDS
  __shared__ float lds[2 * kThreads * kLdsStride];
  const uint32_t slotP = tid;
  const uint32_t slotQ = tid + kThreads;

  // ---- 1. kick off async staging of both pixels' 16 channel values -------
  // Lanes are consecutive columns -> each async op moves one contiguous
  // 128B line per channel plane. DMA overlaps with the threefry/sort below.
#pragma unroll
  for (uint32_t c = 0; c < kC; ++c) {
    const float* gp = img + (size_t)c * kPlane + p;
    const float* gq = img + (size_t)c * kPlane + q;
    ASYNC_COPY_B32(gp, &lds[slotP * kLdsStride + c]);
    ASYNC_COPY_B32(gq, &lds[slotQ * kLdsStride + c]);
  }

  // ---- 2. RNG: 16 threefry blocks serve BOTH pixels (JAX counter split) ---
  // flat noise index n = pixel*16 + c; block b = n (n < N/2) with
  // counter (b, b + N/2); out0 -> pixel p, out1 -> pixel p + 2M.
  // uniform = bitcast(bits>>9 | 0x3f800000) - 1 is monotone in the top 23
  // bits, so key = (bits & ~0x1FF) | c sorts identically to stable argsort
  // (ties broken by channel index in the cleared low bits).
  uint32_t keyA[16], keyB[16];
#pragma unroll
  for (uint32_t c = 0; c < kC; ++c) {
    const uint32_t b = p * kC + c;
    uint32_t o0, o1;
    threefry2x32_key01(b, b + kHalfElem, o0, o1);
    keyA[c] = (o0 & 0xFFFFFE00u) | c;
    keyB[c] = (o1 & 0xFFFFFE00u) | c;
  }
  batcher16(keyA);
  batcher16(keyB);

  // ---- 3. wait for own wave's async copies (no cross-thread sharing) ------
  asm volatile("s_wait_asynccnt 0" ::: "memory");

  // ---- 4. gather through the permutation and stream out (non-temporal) ----
#pragma unroll
  for (uint32_t k = 0; k < kC; ++k) {
    const float va = lds[slotP * kLdsStride + (keyA[k] & 15u)];
    const float vb = lds[slotQ * kLdsStride + (keyB[k] & 15u)];
    __builtin_nontemporal_store(va, out + (size_t)k * kPlane + p);
    __builtin_nontemporal_store(vb, out + (size_t)k * kPlane + q);
  }
}

extern "C" void kernel_launch(void* const* d_in, const int* in_sizes, int n_in,
                              void* d_out, int out_size, void* d_ws, size_t ws_size,
                              hipStream_t stream) {
  (void)in_sizes; (void)n_in; (void)out_size; (void)d_ws; (void)ws_size;
  const float* img = (const float*)d_in[0];
  float* out = (float*)d_out;
  const uint32_t blocks = kHalfPix / kThreads;  // 8192 blocks x 256 threads
  PerPixelChannelPermutation_kernel<<<blocks, kThreads, 0, stream>>>(img, out);
}